// Message_block_25082609009453
// MI455X (gfx1250) — compile-verified
//
#include <hip/hip_runtime.h>
#include <math.h>

typedef __attribute__((ext_vector_type(2))) float v2f;
typedef __attribute__((ext_vector_type(8))) float v8f;

#define D_IN   128
#define D_H    128
#define D_OUT  384
#define LDS_STRIDE 132   // 128 + 4 pad -> bank-conflict-free A-frag reads

__launch_bounds__(256, 1)
__global__ void msg_block_25082609009453_kernel(
    const float* __restrict__ input1, const float* __restrict__ input2,
    const float* __restrict__ v_j,    const float* __restrict__ s_j,
    const float* __restrict__ v_norm, const float* __restrict__ W1,
    const float* __restrict__ b1,     const float* __restrict__ W2,
    const float* __restrict__ b2,     const float* __restrict__ Wr,
    const float* __restrict__ br,     float* __restrict__ out)
{
    __shared__ float h_lds[16 * LDS_STRIDE];
    __shared__ float s_in2[16];
    __shared__ float wave_s[8];
    __shared__ float wave_t[8];
    __shared__ float totals[2];

    const int tid     = threadIdx.x;
    const int wave    = tid >> 5;
    const int lane    = tid & 31;
    const int halfsel = lane >> 4;   // 0 | 1 (lane group)
    const int row16   = lane & 15;   // A-frag M / B,C,D-frag N-in-tile
    const int kb      = halfsel << 1;

    if (tid < 16) s_in2[tid] = input2[tid];   // rows 0..15 of (N,1) input2

    // ---------- GEMM1: h = silu(input1[0:16] @ W1 + b1); 1 N-tile per wave
    {
        const int ncol = wave * 16 + row16;
        v8f acc = {};
        #pragma unroll 4
        for (int k0 = 0; k0 < D_IN; k0 += 4) {
            v2f a, b;
            a.x = input1[row16 * D_IN + k0 + kb];
            a.y = input1[row16 * D_IN + k0 + kb + 1];
            b.x = W1[(k0 + kb)     * D_H + ncol];
            b.y = W1[(k0 + kb + 1) * D_H + ncol];
            acc = __builtin_amdgcn_wmma_f32_16x16x4_f32(
                      false, a, false, b, (short)0, acc, false, false);
        }
        const float bias = b1[ncol];
        #pragma unroll
        for (int v = 0; v < 8; ++v) {
            const int m = v + 8 * halfsel;
            float x = acc[v] + bias;
            h_lds[m * LDS_STRIDE + ncol] = x / (1.0f + __expf(-x)); // silu
        }
    }
    __syncthreads();

    // ---------- GEMM2 + fused epilogue; 3 N-tiles per wave (24 total)
    const float f_c = 1.29389262614623657f;   // 0.5*cos(pi*1.5/5) + 1
    float s_loc = 0.0f, t_loc = 0.0f;
    for (int t = 0; t < 3; ++t) {
        const int ncol = (wave * 3 + t) * 16 + row16;
        v8f acc = {};
        #pragma unroll 4
        for (int k0 = 0; k0 < D_H; k0 += 4) {
            v2f a, b;
            a.x = h_lds[row16 * LDS_STRIDE + k0 + kb];
            a.y = h_lds[row16 * LDS_STRIDE + k0 + kb + 1];
            b.x = W2[(k0 + kb)     * D_OUT + ncol];
            b.y = W2[(k0 + kb + 1) * D_OUT + ncol];
            acc = __builtin_amdgcn_wmma_f32_16x16x4_f32(
                      false, a, false, b, (short)0, acc, false, false);
        }
        const float bias2 = b2[ncol];
        const float wrn   = Wr[ncol];
        const float brn   = br[ncol];
        #pragma unroll
        for (int v = 0; v < 8; ++v) {
            const int m = v + 8 * halfsel;             // output row 0..15
            float phi = acc[v] + bias2;
            float w   = (s_in2[m] * wrn + brn) * f_c;
            float o   = phi * w;
            if (m >= 3 && m <= 5) s_loc += o;                              // s_m rows
            if (m <= 2)           t_loc += o * v_j[m * D_OUT + ncol];      // out1
            if (m >= 6 && m <= 8) t_loc += o * v_norm[(m - 6) * D_OUT + ncol]; // out3
        }
    }

    // ---------- deterministic reduction: wave32 butterfly -> fixed-order sum
    #pragma unroll
    for (int off = 16; off > 0; off >>= 1) {
        s_loc += __shfl_xor(s_loc, off, 32);
        t_loc += __shfl_xor(t_loc, off, 32);
    }
    if (lane == 0) { wave_s[wave] = s_loc; wave_t[wave] = t_loc; }
    __syncthreads();
    if (tid == 0) {
        float ss = 0.0f, tt = 0.0f;
        for (int wv = 0; wv < 8; ++wv) { ss += wave_s[wv]; tt += wave_t[wv]; }
        totals[0] = ss; totals[1] = tt;
    }
    __syncthreads();

    // ---------- outputs: d_out[0:1152] = v_m = sum + v_j ; d_out[1152] = s_m
    const float tt = totals[1];
    for (int i = tid; i < 3 * D_OUT; i += 256)
        out[i] = tt + v_j[i];
    if (tid == 0)
        out[3 * D_OUT] = totals[0] + s_j[0];
}

extern "C" void kernel_launch(void* const* d_in, const int* in_sizes, int n_in,
                              void* d_out, int out_size, void* d_ws, size_t ws_size,
                              hipStream_t stream) {
    const float* input1 = (const float*)d_in[0];
    const float* input2 = (const float*)d_in[1];
    const float* v_j    = (const float*)d_in[2];
    const float* s_j    = (const float*)d_in[3];
    const float* v_norm = (const float*)d_in[4];
    const float* W1     = (const float*)d_in[5];
    const float* b1     = (const float*)d_in[6];
    const float* W2     = (const float*)d_in[7];
    const float* b2     = (const float*)d_in[8];
    const float* Wr     = (const float*)d_in[9];
    const float* br     = (const float*)d_in[10];

    msg_block_25082609009453_kernel<<<1, 256, 0, stream>>>(
        input1, input2, v_j, s_j, v_norm, W1, b1, W2, b2, Wr, br, (float*)d_out);
}